// Coo2FulSimple_13134009991371
// MI455X (gfx1250) — compile-verified
//
#include <hip/hip_runtime.h>

// CDNA5 / gfx1250, wave32.
// Neighbor-matrix kernel: out[b,i,j,s] = masked |r_i - r_j + shift_s|^2, plus mask.
//
// Core trick: sod = |r_i|^2 + |q_c|^2 - 2 r_i . q_c with q_c = r_j - shift_s.
// This is a K=4 bilinear form -> one V_WMMA_F32_16X16X4_F32 per 16x16 tile,
// with |q_c|^2 preloaded in the C accumulator. Kernel is HBM-store bound
// (~226 MB out => ~10 us floor at 23.3 TB/s); compute is negligible.

typedef __attribute__((ext_vector_type(2))) float v2f;
typedef __attribute__((ext_vector_type(8))) float v8f;

#define NB   4            // batches
#define NA   512          // atoms
#define NS   27           // shift candidates
#define NC   (NA * NS)    // flattened (j,s) columns per batch = 13824
#define RC2  36.0f        // cutoff^2

__global__ __launch_bounds__(32)
void coo2ful_wmma_kernel(const float* __restrict__ pos,          // [B,N,3]
                         const float* __restrict__ cel,          // [B,3,3]
                         const unsigned char* __restrict__ ent,  // [B,N] bool
                         const int* __restrict__ sft,            // [S,3]
                         float* __restrict__ out)                // [2, B,N,N,S] (sod, mask)
{
    const int lane = threadIdx.x;      // 0..31, one wave per block
    const int m    = lane & 15;        // tile row (for A) / tile col (for B,C,D)
    const int hi   = lane >> 4;        // half-wave select (K-split / row+8)

    const int ct = blockIdx.x;         // column tile: 0..863
    const int rt = blockIdx.y;         // row tile:    0..31
    const int b  = blockIdx.z;         // batch:       0..3

    const int i0 = rt * 16;
    const int c0 = ct * 16;

    // ---------------- A tile: rows i0..i0+15, A[i] = [-2x, -2y, -2z, |r_i|^2]
    // f32 16x4 A layout: lanes 0-15 hold K=0..1 (VGPR0,VGPR1), lanes 16-31 K=2..3.
    const float* pr = pos + ((size_t)b * NA + (i0 + m)) * 3;
    const float px = pr[0], py = pr[1], pz = pr[2];
    const float nr = px * px + py * py + pz * pz;
    v2f A;
    A.x = hi ? (-2.0f * pz) : (-2.0f * px);
    A.y = hi ? nr           : (-2.0f * py);

    // ---------------- this lane's column c = j*27 + s
    const int col = c0 + m;
    const int j   = col / NS;
    const int s   = col - j * NS;

    const int sx = sft[s * 3 + 0];
    const int sy = sft[s * 3 + 1];
    const int sz = sft[s * 3 + 2];
    const bool zero_sft = ((sx | sy | sz) == 0);

    // cartesian shift: sft_xyz = sft_cel . cel_mat   (row-vector times matrix)
    const float* cb = cel + (size_t)b * 9;
    const float fx = (float)sx, fy = (float)sy, fz = (float)sz;
    const float shx = fx * cb[0] + fy * cb[3] + fz * cb[6];
    const float shy = fx * cb[1] + fy * cb[4] + fz * cb[7];
    const float shz = fx * cb[2] + fy * cb[5] + fz * cb[8];

    const float* pj = pos + ((size_t)b * NA + j) * 3;
    const float qx = pj[0] - shx;
    const float qy = pj[1] - shy;
    const float qz = pj[2] - shz;
    const float nq = qx * qx + qy * qy + qz * qz;

    // ---------------- B tile: B[:,c] = [qx, qy, qz, 1]^T
    // f32 4x16 B layout mirrors A: lanes 0-15 hold K=0..1, lanes 16-31 K=2..3.
    v2f Bm;
    Bm.x = hi ? qz   : qx;
    Bm.y = hi ? 1.0f : qy;

    // ---------------- C tile: |q_c|^2 broadcast down the column (all 8 rows/lane)
    v8f C;
#pragma unroll
    for (int v = 0; v < 8; ++v) C[v] = nq;

    // D = A x B + C  ==  full 16x16 tile of squared distances, f32 exact path.
    // (8 args: neg_a, A, neg_b, B, c_mod, C, reuse_a, reuse_b)
    v8f D = __builtin_amdgcn_wmma_f32_16x16x4_f32(
        false, A, false, Bm, (short)0, C, false, false);

    // ---------------- mask + store (store-bandwidth bound part)
    const bool entj = ent[(size_t)b * NA + j] != 0;
    const size_t maskOfs = (size_t)NB * NA * NC;   // second output (mask) offset

#pragma unroll
    for (int v = 0; v < 8; ++v) {
        const int row  = i0 + v + 8 * hi;          // D row for this VGPR
        const bool enti = ent[(size_t)b * NA + row] != 0;
        const float sod = D[v];
        const bool ok = (sod <= RC2) && enti && entj && !((row == j) && zero_sft);
        const size_t idx = ((size_t)b * NA + row) * (size_t)NC + (size_t)col;
        out[idx]           = ok ? sod : 0.0f;      // masked squared distance
        out[maskOfs + idx] = ok ? 1.0f : 0.0f;     // boolean mask as 0/1
    }
}

extern "C" void kernel_launch(void* const* d_in, const int* in_sizes, int n_in,
                              void* d_out, int out_size, void* d_ws, size_t ws_size,
                              hipStream_t stream)
{
    // setup_inputs() order: pos_xyz[f32], cel_mat[f32], pbc[bool] (unused by the
    // reference math), ent[bool], sft_cel[int32]
    const float*         pos = (const float*)d_in[0];
    const float*         cel = (const float*)d_in[1];
    const unsigned char* ent = (const unsigned char*)d_in[3];
    const int*           sft = (const int*)d_in[4];
    float*               out = (float*)d_out;

    dim3 grid(NC / 16, NA / 16, NB);   // (864, 32, 4) 16x16 tiles, 1 wave each
    coo2ful_wmma_kernel<<<grid, dim3(32), 0, stream>>>(pos, cel, ent, sft, out);
}